// RmseLoss_sum5_35270271434986
// MI455X (gfx1250) — compile-verified
//
#include <hip/hip_runtime.h>
#include <math.h>

// ---------------------------------------------------------------------------
// Shapes fixed by the reference: inputs (T,R,5), yy_* (T,R,1/2), rain (T,R,1)
// ---------------------------------------------------------------------------
#define T_DIM   2000
#define R_DIM   4000
#define NYC     5
#define NROWS   (T_DIM * R_DIM)        // 8,000,000
#define NGROUPS (NROWS / 4)            // 4 rows (20 floats = 5 float4) / thread
#define NCHUNK  32
#define LPAIRS  63                     // ceil((T-1)/NCHUNK) = ceil(1999/32)

typedef __attribute__((ext_vector_type(2))) float v2f;
typedef __attribute__((ext_vector_type(8))) float v8f;

// ---------------------------------------------------------------------------
// Wave-wide (32-lane) sum using V_WMMA_F32_16X16X4_F32.
// A (16x4 f32): lane l supplies A[m = l%16][k = 2*(l/16)] = v, next k = 0.
// B = all ones  -> D[m][n] = v_m + v_{m+16} for every n.
// Lane holds D components for M = vgpr + 8*(l/16); component sum gives
// sum(s_0..s_7) on lanes 0-15 and sum(s_8..s_15) on lanes 16-31; one xor-16
// shuffle-add yields the full 32-lane total on every lane.
// Call only from convergent code (EXEC must be all ones for WMMA).
// ---------------------------------------------------------------------------
__device__ __forceinline__ float wave_sum_wmma(float v) {
    v2f a; a.x = v;    a.y = 0.0f;
    v2f b; b.x = 1.0f; b.y = 1.0f;
    v8f c = {};
    v8f d = __builtin_amdgcn_wmma_f32_16x16x4_f32(
        /*neg_a=*/false, a, /*neg_b=*/false, b,
        /*c_mod=*/(short)0, c, /*reuse_a=*/false, /*reuse_b=*/false);
    float t = 0.0f;
#pragma unroll
    for (int i = 0; i < 8; ++i) t += d[i];
    t += __shfl_xor(t, 16, 32);
    return t;
}

// ---------------------------------------------------------------------------
// ws double accumulators (16):
//   [0..4]  sum of masked (inputs-targets)^2 per channel
//   [5..9]  non-NaN counts per channel
//   [10] s1  [11] n1  [12] s2  [13] n2
//   [14] loss3 sum-of-segment-means   [15] loss3 segment count
// ---------------------------------------------------------------------------
__global__ void init_acc_kernel(double* __restrict__ acc) {
    int i = threadIdx.x;
    if (i < 16) acc[i] = 0.0;
}

// ---------------------------------------------------------------------------
// Streaming kernel: loss0/1/2 partials over 4 rows / 20 floats per iteration.
// 20 floats = 5 aligned float4 loads, and every element's channel (e % 5) is
// a compile-time constant, so accumulation is fully unrolled & branchless.
// Also compacts inputs[:,:,0] (coalesced float4 store) for the loss3 pass.
// ---------------------------------------------------------------------------
__global__ __launch_bounds__(256)
void loss012_kernel(const float4* __restrict__ inputs4,
                    const float4* __restrict__ targets4,
                    const float4* __restrict__ yy_p4,
                    const float4* __restrict__ yy_train4,
                    float4* __restrict__ compact4,   // may be null
                    double* __restrict__ acc) {
    float a[14];
#pragma unroll
    for (int j = 0; j < 14; ++j) a[j] = 0.0f;

    const long long stride = (long long)gridDim.x * blockDim.x;
    for (long long g = (long long)blockIdx.x * blockDim.x + threadIdx.x;
         g < NGROUPS; g += stride) {
        // ---- 20 input / 20 target floats (rows 4g..4g+3) ----
        float iv[20], tv[20];
#pragma unroll
        for (int q = 0; q < 5; ++q) {
            float4 fi = inputs4[5 * g + q];
            float4 ft = targets4[5 * g + q];
            iv[4 * q + 0] = fi.x; iv[4 * q + 1] = fi.y;
            iv[4 * q + 2] = fi.z; iv[4 * q + 3] = fi.w;
            tv[4 * q + 0] = ft.x; tv[4 * q + 1] = ft.y;
            tv[4 * q + 2] = ft.z; tv[4 * q + 3] = ft.w;
        }
#pragma unroll
        for (int e = 0; e < 20; ++e) {
            const int ch = e % 5;                 // compile-time constant
            float t  = tv[e];
            float m  = (t == t) ? 1.0f : 0.0f;    // !isnan, branchless
            float dd = m * (iv[e] - (m != 0.0f ? t : 0.0f));
            a[ch]     += dd * dd;
            a[5 + ch] += m;
        }
        if (compact4) {
            float4 c0; c0.x = iv[0]; c0.y = iv[5]; c0.z = iv[10]; c0.w = iv[15];
            compact4[g] = c0;                     // channel-0 compaction
        }

        // ---- yy_p (4 floats) and yy_train (8 floats) for rows 4g..4g+3 ----
        float4 p4 = yy_p4[g];
        float4 t0 = yy_train4[2 * g];
        float4 t1 = yy_train4[2 * g + 1];
        float p[4]    = { p4.x, p4.y, p4.z, p4.w };
        float val[4]  = { t0.x, t0.z, t1.x, t1.z };
        float mark[4] = { t0.y, t0.w, t1.y, t1.w };
#pragma unroll
        for (int r = 0; r < 4; ++r) {
            float m1 = (mark[r] == 1.0f) ? 1.0f : 0.0f;
            float m2 = (mark[r] == 2.0f) ? 1.0f : 0.0f;
            float e1 = p[r] - val[r];
            a[10] += m1 * e1 * e1;  a[11] += m1;
            a[12] += m2 * p[r] * p[r];  a[13] += m2;
        }
    }

    // wave reduction via WMMA, then cross-wave via LDS, one f64 atomic each
    __shared__ double lds[8][14];
    const int wave = threadIdx.x >> 5;
    const int lane = threadIdx.x & 31;
#pragma unroll
    for (int j = 0; j < 14; ++j) {
        float w = wave_sum_wmma(a[j]);
        if (lane == 0) lds[wave][j] = (double)w;
    }
    __syncthreads();
    if (threadIdx.x < 14) {
        double s = 0.0;
#pragma unroll
        for (int w = 0; w < 8; ++w) s += lds[w][threadIdx.x];
        atomicAdd(&acc[threadIdx.x], s);
    }
}

// ---------------------------------------------------------------------------
// loss3 pass 1: per (column, time-chunk) run summary.
// A "segment" with nonzero count == maximal run of consecutive pairs t with
// rain[t]==rain[t-1]. Summary: {allOpen, preS, preC, midMeanSum, midN,
//                               sufS, sufC, npairs}
// ---------------------------------------------------------------------------
__global__ __launch_bounds__(256)
void seg_pass1_kernel(const float* __restrict__ inputs,   // strided fallback
                      const float* __restrict__ compact0, // coalesced path
                      const float* __restrict__ rain,
                      float4* __restrict__ summ) {
    int tid = blockIdx.x * blockDim.x + threadIdx.x;
    if (tid >= R_DIM * NCHUNK) return;
    const int r     = tid % R_DIM;
    const int chunk = tid / R_DIM;
    const int p0 = chunk * LPAIRS + 1;
    const int p1 = min((chunk + 1) * LPAIRS, T_DIM - 1);
    float4* out = summ + (size_t)tid * 2;
    if (p0 > T_DIM - 1) {
        float4 z; z.x = z.y = z.z = z.w = 0.0f;
        out[0] = z; out[1] = z;
        return;
    }

    const size_t idx0 = (size_t)(p0 - 1) * R_DIM + r;
    float dprev = compact0 ? compact0[idx0] : inputs[idx0 * NYC];
    float rprev = rain[idx0];

    float runS = 0.f, runC = 0.f, preS = 0.f, preC = 0.f;
    float midMean = 0.f, midN = 0.f;
    bool preClosed = false;
    for (int t = p0; t <= p1; ++t) {
        const size_t idx = (size_t)t * R_DIM + r;
        float dc = compact0 ? compact0[idx] : inputs[idx * NYC];
        float rc = rain[idx];
        float df = dc - dprev;
        float sq = df * df;
        if (rc == rprev) { runS += sq; runC += 1.0f; }
        else {
            if (!preClosed)       { preS = runS; preC = runC; preClosed = true; }
            else if (runC > 0.f)  { midMean += runS / runC; midN += 1.0f; }
            runS = 0.f; runC = 0.f;
        }
        dprev = dc; rprev = rc;
    }
    float allOpen = preClosed ? 0.0f : 1.0f;
    if (!preClosed) { preS = runS; preC = runC; runS = 0.f; runC = 0.f; }
    float4 o0; o0.x = allOpen; o0.y = preS; o0.z = preC; o0.w = midMean;
    float4 o1; o1.x = midN;    o1.y = runS; o1.z = runC; o1.w = (float)(p1 - p0 + 1);
    out[0] = o0; out[1] = o1;
}

// ---------------------------------------------------------------------------
// loss3 pass 2: stitch chunk summaries per column.
// Runs merge across a chunk boundary iff the trailing pair of the left chunk
// and the leading pair of the right chunk are both "same-sign" pairs
// (carryC>0 && preC>0), since boundary pairs are adjacent by construction.
// ---------------------------------------------------------------------------
__global__ __launch_bounds__(256)
void seg_pass2_kernel(const float4* __restrict__ summ, double* __restrict__ acc) {
    int r = blockIdx.x * blockDim.x + threadIdx.x;
    if (r >= R_DIM) return;
    float carryS = 0.f, carryC = 0.f;
    double meanSum = 0.0, nRun = 0.0;
    for (int c = 0; c < NCHUNK; ++c) {
        const float4* s = summ + ((size_t)c * R_DIM + r) * 2;
        float4 s0 = s[0], s1 = s[1];
        float allOpen = s0.x, preS = s0.y, preC = s0.z, midMean = s0.w;
        float midN = s1.x, sufS = s1.y, sufC = s1.z, np = s1.w;
        if (np <= 0.f) continue;
        if (allOpen != 0.f) {                 // whole chunk one open run
            carryS += preS; carryC += preC;   // (carry may be empty: fine)
        } else {
            if (preC > 0.f) {                 // close carry+prefix as one run
                float rs = carryS + preS, rc = carryC + preC;
                meanSum += (double)(rs / rc); nRun += 1.0;
            } else if (carryC > 0.f) {        // first pair breaks the carry
                meanSum += (double)(carryS / carryC); nRun += 1.0;
            }
            carryS = 0.f; carryC = 0.f;
            meanSum += (double)midMean; nRun += (double)midN;
            if (sufC > 0.f) { carryS = sufS; carryC = sufC; }
        }
    }
    if (carryC > 0.f) { meanSum += (double)(carryS / carryC); nRun += 1.0; }
    atomicAdd(&acc[14], meanSum);
    atomicAdd(&acc[15], nRun);
}

// ---------------------------------------------------------------------------
// Finalize: combine accumulators into the scalar loss.
// ---------------------------------------------------------------------------
__global__ void finalize_kernel(const double* __restrict__ acc,
                                float* __restrict__ out) {
    if (threadIdx.x == 0 && blockIdx.x == 0) {
        double loss0 = 0.0;
        for (int ch = 0; ch < NYC; ++ch)
            loss0 += sqrt(acc[ch] / fmax(acc[5 + ch], 1.0));
        double n1 = acc[11], n2 = acc[13], nR = acc[15];
        double loss1 = (n1 > 0.0) ? sqrt(acc[10] / fmax(n1, 1.0)) : 0.0;
        double loss2 = (n2 > 0.0) ? sqrt(acc[12] / fmax(n2, 1.0)) : 0.0;
        double loss3 = (nR > 0.0) ? (acc[14] / fmax(nR, 1.0)) : 0.0;
        out[0] = (float)(0.6 * loss0 + 0.1 * loss1 + 0.1 * loss2 + 0.1 * loss3);
    }
}

// ---------------------------------------------------------------------------
extern "C" void kernel_launch(void* const* d_in, const int* in_sizes, int n_in,
                              void* d_out, int out_size, void* d_ws, size_t ws_size,
                              hipStream_t stream) {
    const float* inputs   = (const float*)d_in[0];
    const float* targets  = (const float*)d_in[1];
    const float* yy_p     = (const float*)d_in[2];
    const float* yy_train = (const float*)d_in[3];
    const float* rain     = (const float*)d_in[4];
    (void)in_sizes; (void)n_in; (void)out_size;

    // ws layout: [0,256) f64 accumulators | optional 32MB channel-0 compact |
    //            chunk summaries (NCHUNK*R*8 floats ~= 4MB)
    const size_t ACC_BYTES     = 256;
    const size_t COMPACT_BYTES = (size_t)NROWS * sizeof(float);           // 32 MB
    const size_t SUMM_BYTES    = (size_t)NCHUNK * R_DIM * 8 * sizeof(float);

    double* acc = (double*)d_ws;
    bool useCompact = ws_size >= ACC_BYTES + COMPACT_BYTES + SUMM_BYTES;
    float* compact0 = useCompact ? (float*)((char*)d_ws + ACC_BYTES) : nullptr;
    float4* summ    = (float4*)((char*)d_ws + ACC_BYTES +
                                (useCompact ? COMPACT_BYTES : 0));

    init_acc_kernel<<<1, 32, 0, stream>>>(acc);

    loss012_kernel<<<2048, 256, 0, stream>>>(
        (const float4*)inputs, (const float4*)targets,
        (const float4*)yy_p, (const float4*)yy_train,
        (float4*)compact0, acc);

    const int p1_threads = R_DIM * NCHUNK;   // 128,000
    seg_pass1_kernel<<<(p1_threads + 255) / 256, 256, 0, stream>>>(
        inputs, compact0, rain, summ);

    seg_pass2_kernel<<<(R_DIM + 255) / 256, 256, 0, stream>>>(summ, acc);

    finalize_kernel<<<1, 32, 0, stream>>>(acc, (float*)d_out);
}